// PredNet_67293547593778
// MI455X (gfx1250) — compile-verified
//
#include <hip/hip_runtime.h>
#include <hip/hip_bf16.h>

// ---------------------------------------------------------------------------
// PredNet (multi-community GIN + ensemble heads) for MI455X / gfx1250.
//
// Live subgraph only: hs[2] (concat of community layer-2 outputs) and hs[3]
// (rep GIN) feed the kept heads (preds[-2:]); encoder + heads 0/1 skipped.
//
// Key transforms vs naive:
//  * GIN algebra: relu((h + S h)@W1 + b1) == relu(p + S p + b1), p = h@W1.
//    -> GEMM before scatter: layer-0 scatter is 64-wide (not 128), and the
//       relu(p+agg+b1) is fused into the A-operand of the W2 GEMM.
//  * WMMA GEMM: 8-wave blocks, 128x64 tile/block, 16x64 tile/wave (4 accs,
//    A-fragment reuse x4), weight K-slabs staged in LDS via
//    GLOBAL_LOAD_ASYNC_TO_LDS_B128 (+ s_wait_asynccnt), B frags from LDS.
//  * fp32 WMMA (V_WMMA_F32_16X16X4_F32): exact-precision vs fp32 reference;
//    the workload is scatter/atomic-bound, so low-precision WMMA buys nothing.
//  * Segment sums: unsafeAtomicAdd -> global_atomic_add_f32 (L2 atomics).
//
// Input layout assumption (harness flattens setup_inputs() dict):
//   [0] x (32768x128 f32)      [1] z (32768x256 f32)
//   params flattened with sorted dict keys: 'com','enc','pred','rep';
//   each MLP dict in sorted-key order: W1, W2, b1, b2, be, g.
//     com[k][l]  -> idx 2  + (k*2+l)*6 + field      (48 arrays)
//     enc        -> idx 50 (W), 51 (b)              (unused)
//     pred[i]    -> idx 52 + i*6 + field            (24 arrays)
//     rep[0]     -> idx 76 + field                  (6 arrays)
//   [82] edge_index (2x524288 i32)   [83] batch (32768 i32)
// ---------------------------------------------------------------------------

#define N_NODES   32768
#define N_EDGES   524288
#define IN_DIM    128
#define EMB_DIM   256
#define N_COMS    4
#define COM_DIM   64
#define N_CLASSES 10
#define N_GRAPHS  256
#define BN_EPS    1e-5f

typedef __attribute__((ext_vector_type(2))) float v2f;
typedef __attribute__((ext_vector_type(8))) float v8f;
typedef __attribute__((ext_vector_type(4))) int   v4i;

#if __has_builtin(__builtin_amdgcn_global_load_async_to_lds_b128) && \
    __has_builtin(__builtin_amdgcn_s_wait_asynccnt)
#define USE_ASYNC_LDS 1
#else
#define USE_ASYNC_LDS 0
#endif

// ------------------------------- utility -----------------------------------

__global__ void k_zero(float* __restrict__ p, long n) {
  long i = (long)blockIdx.x * blockDim.x + threadIdx.x;
  if (i < n) p[i] = 0.f;
}

// -------------------------- edge softmax weights ----------------------------
__global__ void k_edge_softmax(const float* __restrict__ z,
                               const int* __restrict__ ei,
                               float* __restrict__ ew) {
  int e = blockIdx.x * blockDim.x + threadIdx.x;
  if (e >= N_EDGES) return;
  int s = ei[e], d = ei[N_EDGES + e];
  const float4* zs = (const float4*)(z + (long)s * EMB_DIM);
  const float4* zd = (const float4*)(z + (long)d * EMB_DIM);
  float dot[N_COMS];
  float m = -1e30f;
#pragma unroll
  for (int k = 0; k < N_COMS; ++k) {
    float acc = 0.f;
#pragma unroll
    for (int j = 0; j < COM_DIM / 4; ++j) {
      float4 a = zs[k * (COM_DIM / 4) + j];
      float4 b = zd[k * (COM_DIM / 4) + j];
      acc += a.x * b.x + a.y * b.y + a.z * b.z + a.w * b.w;
    }
    dot[k] = acc;                     // TAU == 1.0
    m = fmaxf(m, acc);
  }
  float sum = 0.f;
#pragma unroll
  for (int k = 0; k < N_COMS; ++k) { dot[k] = __expf(dot[k] - m); sum += dot[k]; }
  float inv = 1.f / sum;
#pragma unroll
  for (int k = 0; k < N_COMS; ++k) ew[(long)e * N_COMS + k] = dot[k] * inv;
}

// --------------------------- weighted scatter-add ---------------------------
// agg[dst[e], c:c+4] += h[src[e], hoff+c : hoff+c+4] * w(e)
__global__ void k_scatter_add(const float* __restrict__ h, int ldh, int hoff, int d,
                              const int* __restrict__ ei,
                              const float* __restrict__ ew, int ewOff,  // null -> 1
                              float* __restrict__ agg) {
  long t = (long)blockIdx.x * blockDim.x + threadIdx.x;
  int vec = d >> 2;
  long total = (long)N_EDGES * vec;
  if (t >= total) return;
  int e  = (int)(t / vec);
  int c4 = (int)(t % vec) * 4;
  int s = ei[e], dst = ei[N_EDGES + e];
  float w = ew ? ew[(long)e * N_COMS + ewOff] : 1.f;
  float4 hv = *(const float4*)(h + (long)s * ldh + hoff + c4);
  float* ap = agg + (long)dst * d + c4;
  unsafeAtomicAdd(ap + 0, hv.x * w);
  unsafeAtomicAdd(ap + 1, hv.y * w);
  unsafeAtomicAdd(ap + 2, hv.z * w);
  unsafeAtomicAdd(ap + 3, hv.w * w);
}

// ------------------------------ WMMA GEMM -----------------------------------
// C[M,Nn] = (PRE ? relu(A + A2 + preb) : A) @ W + bias
// Block = 256 thr (8 wave32) -> 128x64 tile; wave -> 16x64 (4 WMMA tiles).
// Weight K-slab (32x64) staged to LDS (async-to-LDS when available).
// V_WMMA_F32_16X16X4_F32 fragments (ISA 7.12.2):
//   A 16x4 : lane<16 -> row=lane, K={k,k+1}; lane>=16 -> K={k+2,k+3}
//   B 4x16 : lane<16 -> col=lane, K={k,k+1}; lane>=16 -> col=lane-16, K={k+2,k+3}
//   C 16x16: vgpr j  -> M=j+8*(lane>=16), N=lane&15
#define KS  32           // K-slab
#define WST 68           // LDS row stride in floats (272B: 16B-aligned rows)

template <int PRE>
__global__ void __launch_bounds__(256)
k_gemm(const float* __restrict__ A, int lda, int aoff,
       const float* __restrict__ A2, int lda2, const float* __restrict__ preb,
       const float* __restrict__ W, const float* __restrict__ bias,
       float* __restrict__ C, int ldc, int M, int Nn, int K) {
  __shared__ float sW[KS * WST];
  const int tid  = threadIdx.x;
  const int wave = tid >> 5, lane = tid & 31;
  const int half = lane >> 4, l15 = lane & 15;
  const int panelsN = Nn >> 6;
  const int pM = blockIdx.x / panelsN, pN = blockIdx.x % panelsN;
  const int row   = pM * 128 + wave * 16 + l15;
  const int nBase = pN * 64;

  const float* aRow  = A + (long)row * lda + aoff;
  const float* aRow2 = (PRE == 1) ? (A2 + (long)row * lda2) : nullptr;

  v8f acc0 = {}, acc1 = {}, acc2 = {}, acc3 = {};

  for (int k0 = 0; k0 < K; k0 += KS) {
    // stage W[k0:k0+KS, nBase:nBase+64] -> LDS (512 float4 xfers, 2/thread)
    for (int i = tid; i < KS * 16; i += 256) {
      int r = i >> 4, c4 = (i & 15) << 2;
      const float* gp = W + (long)(k0 + r) * Nn + nBase + c4;
      float* lp = &sW[r * WST + c4];
#if USE_ASYNC_LDS
      __builtin_amdgcn_global_load_async_to_lds_b128(
          (__attribute__((address_space(1))) v4i*)gp,
          (__attribute__((address_space(3))) v4i*)lp, 0, 0);
#else
      *(float4*)lp = *(const float4*)gp;
#endif
    }
#if USE_ASYNC_LDS
    __builtin_amdgcn_s_wait_asynccnt(0);
#endif
    __syncthreads();

#pragma unroll
    for (int kk = 0; kk < KS; kk += 4) {
      const int ka = kk + 2 * half;
      v2f a = *(const v2f*)(aRow + k0 + ka);
      if (PRE == 1) {
        a += *(const v2f*)(aRow2 + k0 + ka);
        a += *(const v2f*)(preb + k0 + ka);
        a.x = fmaxf(a.x, 0.f);
        a.y = fmaxf(a.y, 0.f);
      }
      v2f b;
      b.x = sW[ka * WST + l15];        b.y = sW[(ka + 1) * WST + l15];
      acc0 = __builtin_amdgcn_wmma_f32_16x16x4_f32(false, a, false, b, (short)0, acc0, false, false);
      b.x = sW[ka * WST + 16 + l15];   b.y = sW[(ka + 1) * WST + 16 + l15];
      acc1 = __builtin_amdgcn_wmma_f32_16x16x4_f32(false, a, false, b, (short)0, acc1, false, false);
      b.x = sW[ka * WST + 32 + l15];   b.y = sW[(ka + 1) * WST + 32 + l15];
      acc2 = __builtin_amdgcn_wmma_f32_16x16x4_f32(false, a, false, b, (short)0, acc2, false, false);
      b.x = sW[ka * WST + 48 + l15];   b.y = sW[(ka + 1) * WST + 48 + l15];
      acc3 = __builtin_amdgcn_wmma_f32_16x16x4_f32(false, a, false, b, (short)0, acc3, false, false);
    }
    __syncthreads();
  }

  const int mBase = pM * 128 + wave * 16 + 8 * half;
  v8f accs[4] = { acc0, acc1, acc2, acc3 };
#pragma unroll
  for (int t = 0; t < 4; ++t) {
    const int col = nBase + t * 16 + l15;
    const float bv = bias ? bias[col] : 0.f;
#pragma unroll
    for (int j = 0; j < 8; ++j)
      C[(long)(mBase + j) * ldc + col] = accs[t][j] + bv;
  }
}

// ------------------------------ batchnorm -----------------------------------
__global__ void k_bn_stats(const float* __restrict__ X, int ld, int rows,
                           float* __restrict__ mu, float* __restrict__ var) {
  int f = blockIdx.x;
  float s = 0.f, s2 = 0.f;
  for (int r = threadIdx.x; r < rows; r += blockDim.x) {
    float v = X[(long)r * ld + f];
    s += v; s2 += v * v;
  }
  __shared__ float sh[256], sh2[256];
  sh[threadIdx.x] = s; sh2[threadIdx.x] = s2;
  __syncthreads();
  for (int o = blockDim.x >> 1; o > 0; o >>= 1) {
    if (threadIdx.x < (unsigned)o) { sh[threadIdx.x] += sh[threadIdx.x + o];
                                     sh2[threadIdx.x] += sh2[threadIdx.x + o]; }
    __syncthreads();
  }
  if (threadIdx.x == 0) {
    float m = sh[0] / rows;
    mu[f]  = m;
    var[f] = sh2[0] / rows - m * m;
  }
}

__global__ void k_bn_relu(const float* __restrict__ X, int ld, int d, int rows,
                          const float* __restrict__ g, const float* __restrict__ be,
                          const float* __restrict__ mu, const float* __restrict__ var,
                          float* __restrict__ Y, int ldy, int yoff) {
  long t = (long)blockIdx.x * blockDim.x + threadIdx.x;
  if (t >= (long)rows * d) return;
  int r = (int)(t / d), f = (int)(t % d);
  float v = g[f] * (X[(long)r * ld + f] - mu[f]) * rsqrtf(var[f] + BN_EPS) + be[f];
  Y[(long)r * ldy + yoff + f] = fmaxf(v, 0.f);
}

// ------------------------------ pooling / heads -----------------------------
__global__ void k_pool(const float* __restrict__ H, int ld,
                       const int* __restrict__ batch, float* __restrict__ hg) {
  long t = (long)blockIdx.x * blockDim.x + threadIdx.x;
  if (t >= (long)N_NODES * EMB_DIM) return;
  int r = (int)(t / EMB_DIM), f = (int)(t % EMB_DIM);
  unsafeAtomicAdd(&hg[(long)batch[r] * EMB_DIM + f], H[(long)r * ld + f]);
}

__global__ void k_head_out(const float* __restrict__ U, const float* __restrict__ W2,
                           const float* __restrict__ b2, float* __restrict__ out,
                           int accumulate) {
  int t = blockIdx.x * blockDim.x + threadIdx.x;
  if (t >= N_GRAPHS * N_CLASSES) return;
  int g = t / N_CLASSES, c = t % N_CLASSES;
  float s = b2[c];
  for (int j = 0; j < EMB_DIM; ++j) s += U[g * EMB_DIM + j] * W2[j * N_CLASSES + c];
  if (accumulate) out[t] += s; else out[t] = s;
}

__global__ void k_log_softmax(const float* __restrict__ logits, float* __restrict__ out) {
  int g = blockIdx.x * blockDim.x + threadIdx.x;
  if (g >= N_GRAPHS) return;
  float m = -1e30f;
  for (int c = 0; c < N_CLASSES; ++c) m = fmaxf(m, logits[g * N_CLASSES + c]);
  float s = 0.f;
  for (int c = 0; c < N_CLASSES; ++c) s += __expf(logits[g * N_CLASSES + c] - m);
  float ls = __logf(s) + m;
  for (int c = 0; c < N_CLASSES; ++c) out[g * N_CLASSES + c] = logits[g * N_CLASSES + c] - ls;
}

// ------------------------------ orchestration -------------------------------

static inline dim3 gridFor(long n) { return dim3((unsigned)((n + 255) / 256)); }
static inline dim3 gemmGrid(int M, int Nn) {           // 128x64 per block
  return dim3((unsigned)((long)(M / 128) * (Nn / 64)));
}

extern "C" void kernel_launch(void* const* d_in, const int* in_sizes, int n_in,
                              void* d_out, int out_size, void* d_ws, size_t ws_size,
                              hipStream_t stream) {
  const float* x = (const float*)d_in[0];
  const float* z = (const float*)d_in[1];
  const int P0 = 2;
  auto comP  = [&](int k, int l, int f) { return (const float*)d_in[P0 + (k * 2 + l) * 6 + f]; };
  auto predP = [&](int i, int f)        { return (const float*)d_in[P0 + 50 + i * 6 + f]; };
  auto repP  = [&](int f)               { return (const float*)d_in[P0 + 74 + f]; };
  const int* ei    = (const int*)d_in[P0 + 80];
  const int* batch = (const int*)d_in[P0 + 81];
  // field order per sorted MLP dict: 0=W1 1=W2 2=b1 3=b2 4=be 5=g

  // workspace carve-up (fp32)
  float* ws    = (float*)d_ws;
  float* ew    = ws;                              // N_EDGES*4
  float* agg   = ew    + (long)N_EDGES * 4;       // N*256
  float* pbuf  = agg   + (long)N_NODES * 256;     // N*256  (p = h@W1)
  float* raw   = pbuf  + (long)N_NODES * 256;     // N*256
  float* h1cat = raw   + (long)N_NODES * 256;     // N*256
  float* h2cat = h1cat + (long)N_NODES * 256;     // N*256
  float* h3    = h2cat + (long)N_NODES * 256;     // N*256
  float* hg    = h3    + (long)N_NODES * 256;     // 256*256
  float* praw  = hg    + 256 * 256;               // 256*256
  float* pu    = praw  + 256 * 256;               // 256*256
  float* mu    = pu    + 256 * 256;               // 256
  float* var   = mu    + 256;                     // 256
  float* logit = var   + 256;                     // 256*10

  const dim3 B(256);

  // 1) edge softmax weights
  k_edge_softmax<<<gridFor(N_EDGES), B, 0, stream>>>(z, ei, ew);

  // 2) community GIN towers: GEMM-first formulation
  for (int k = 0; k < N_COMS; ++k) {
    const float* hin[2]  = { x, h1cat };
    const int    ldin[2] = { IN_DIM, EMB_DIM };
    const int    ofin[2] = { 0, COM_DIM * k };
    const int    kin[2]  = { IN_DIM, COM_DIM };
    float*       hout[2] = { h1cat, h2cat };
    for (int l = 0; l < 2; ++l) {
      // p = h @ W1
      k_gemm<0><<<gemmGrid(N_NODES, COM_DIM), B, 0, stream>>>(
          hin[l], ldin[l], ofin[l], nullptr, 0, nullptr,
          comP(k, l, 0), nullptr, pbuf, COM_DIM, N_NODES, COM_DIM, kin[l]);
      // agg = S_k p   (64-wide scatter)
      k_zero<<<gridFor((long)N_NODES * COM_DIM), B, 0, stream>>>(agg, (long)N_NODES * COM_DIM);
      k_scatter_add<<<gridFor((long)N_EDGES * (COM_DIM / 4)), B, 0, stream>>>(
          pbuf, COM_DIM, 0, COM_DIM, ei, ew, k, agg);
      // raw = relu(p + agg + b1) @ W2 + b2
      k_gemm<1><<<gemmGrid(N_NODES, COM_DIM), B, 0, stream>>>(
          pbuf, COM_DIM, 0, agg, COM_DIM, comP(k, l, 2),
          comP(k, l, 1), comP(k, l, 3), raw, COM_DIM, N_NODES, COM_DIM, COM_DIM);
      // bn + relu -> concat slot
      k_bn_stats<<<COM_DIM, B, 0, stream>>>(raw, COM_DIM, N_NODES, mu, var);
      k_bn_relu<<<gridFor((long)N_NODES * COM_DIM), B, 0, stream>>>(
          raw, COM_DIM, COM_DIM, N_NODES, comP(k, l, 5), comP(k, l, 4), mu, var,
          hout[l], EMB_DIM, COM_DIM * k);
    }
  }

  // 3) rep GIN: h2cat(256) -> h3(256), unweighted
  k_gemm<0><<<gemmGrid(N_NODES, EMB_DIM), B, 0, stream>>>(
      h2cat, EMB_DIM, 0, nullptr, 0, nullptr,
      repP(0), nullptr, pbuf, EMB_DIM, N_NODES, EMB_DIM, EMB_DIM);
  k_zero<<<gridFor((long)N_NODES * EMB_DIM), B, 0, stream>>>(agg, (long)N_NODES * EMB_DIM);
  k_scatter_add<<<gridFor((long)N_EDGES * (EMB_DIM / 4)), B, 0, stream>>>(
      pbuf, EMB_DIM, 0, EMB_DIM, ei, nullptr, 0, agg);
  k_gemm<1><<<gemmGrid(N_NODES, EMB_DIM), B, 0, stream>>>(
      pbuf, EMB_DIM, 0, agg, EMB_DIM, repP(2),
      repP(1), repP(3), raw, EMB_DIM, N_NODES, EMB_DIM, EMB_DIM);
  k_bn_stats<<<EMB_DIM, B, 0, stream>>>(raw, EMB_DIM, N_NODES, mu, var);
  k_bn_relu<<<gridFor((long)N_NODES * EMB_DIM), B, 0, stream>>>(
      raw, EMB_DIM, EMB_DIM, N_NODES, repP(5), repP(4), mu, var, h3, EMB_DIM, 0);

  // 4) prediction heads on hs[2]=h2cat (pred[2]) and hs[3]=h3 (pred[3])
  const float* headH[2] = { h2cat, h3 };
  for (int hI = 0; hI < 2; ++hI) {
    int pi = 2 + hI;
    k_zero<<<gridFor((long)N_GRAPHS * EMB_DIM), B, 0, stream>>>(hg, (long)N_GRAPHS * EMB_DIM);
    k_pool<<<gridFor((long)N_NODES * EMB_DIM), B, 0, stream>>>(headH[hI], EMB_DIM, batch, hg);
    k_gemm<0><<<gemmGrid(N_GRAPHS, EMB_DIM), B, 0, stream>>>(
        hg, EMB_DIM, 0, nullptr, 0, nullptr,
        predP(pi, 0), predP(pi, 2), praw, EMB_DIM, N_GRAPHS, EMB_DIM, EMB_DIM);
    k_bn_stats<<<EMB_DIM, B, 0, stream>>>(praw, EMB_DIM, N_GRAPHS, mu, var);
    k_bn_relu<<<gridFor((long)N_GRAPHS * EMB_DIM), B, 0, stream>>>(
        praw, EMB_DIM, EMB_DIM, N_GRAPHS, predP(pi, 5), predP(pi, 4), mu, var,
        pu, EMB_DIM, 0);
    k_head_out<<<gridFor(N_GRAPHS * N_CLASSES), B, 0, stream>>>(
        pu, predP(pi, 1), predP(pi, 3), logit, /*accumulate=*/hI);
  }

  // 5) log-softmax into d_out (256 x 10, fp32)
  k_log_softmax<<<gridFor(N_GRAPHS), B, 0, stream>>>(logit, (float*)d_out);
}